// AttentionPositionAlign_61572651156026
// MI455X (gfx1250) — compile-verified
//
#include <hip/hip_runtime.h>

// Problem sizes (compile-time constants from the reference)
#define B_  8
#define M_  1024
#define N_  36
#define H_  512
#define DQ_ 512
#define DM_ 2048

typedef __attribute__((ext_vector_type(2))) float v2f;
typedef __attribute__((ext_vector_type(8))) float v8f;

// ---------------------------------------------------------------------------
// Kernel 1: tgt[b,n,h] = memory[b,n,:] @ Wm[:,h]   ([288 x 2048] @ [2048 x 512])
// One 16x16 output tile per wave via V_WMMA_F32_16X16X4_F32.
// 288 rows = 18 row-tiles, 512 cols = 32 col-tiles -> 576 wave-tiles
// 8 waves / block -> 72 blocks.
// ---------------------------------------------------------------------------
__global__ __launch_bounds__(256) void tgt_proj_kernel(
    const float* __restrict__ mem, const float* __restrict__ Wm,
    float* __restrict__ tgt)
{
    const int lane = threadIdx.x & 31;
    const int hl   = lane >> 4;     // half-wave: 0 or 1
    const int lm   = lane & 15;     // lane-in-half
    const int wid  = threadIdx.x >> 5;
    const int tile = blockIdx.x * 8 + wid;      // 0..575
    const int rt   = tile >> 5;                 // row tile 0..17
    const int ct   = tile & 31;                 // col tile 0..31
    const int h0   = ct * 16;

    // A rows: memory is [B,N,DM] contiguous -> flat row index rt*16+lm
    const float* arow = mem + (size_t)(rt * 16 + lm) * DM_;

    v8f acc = {};
    for (int k = 0; k < DM_; k += 4) {
        // A frag: lane holds A[row, k+2*hl .. k+2*hl+1] (contiguous -> b64 load)
        v2f a = *(const v2f*)(arow + k + 2 * hl);
        // B frag: lane holds B[k+2*hl, h0+lm] and B[k+2*hl+1, h0+lm]
        const float* bp = Wm + (size_t)(k + 2 * hl) * H_ + h0 + lm;
        v2f bb;
        bb.x = bp[0];
        bb.y = bp[H_];
        acc = __builtin_amdgcn_wmma_f32_16x16x4_f32(
            false, a, false, bb, (short)0, acc, false, false);
    }

    // D layout: VGPR v -> row (v + 8*hl), col h0+lm
#pragma unroll
    for (int v = 0; v < 8; ++v) {
        tgt[(size_t)(rt * 16 + v + 8 * hl) * H_ + h0 + lm] = acc[v];
    }
}

// ---------------------------------------------------------------------------
// Kernel 2 (fused): per workgroup = (batch b, 16 m-rows).
//  Phase 1: S[16][512] = query[b, m0:m0+16, :] @ Wq  (WMMA f32, 8 waves x 4 h-tiles)
//  Phase 2: out[b,m,c] = sum_{n,h} relu(S[m][h] + tgt[b][n][h]) * Wout[n*H+h][c]
// The 604 MB [B,M,N,H] intermediate never exists.
// ---------------------------------------------------------------------------
__global__ __launch_bounds__(256) void fused_align_kernel(
    const float* __restrict__ query, const float* __restrict__ Wq,
    const float* __restrict__ tgt,   const float* __restrict__ Wout,
    float* __restrict__ out)
{
    __shared__ float S[16][H_];         // 32 KB: src projection tile
    __shared__ float red[16][16][4];    //  4 KB: partial-sum reduction

    const int t    = threadIdx.x;
    const int lane = t & 31;
    const int hl   = lane >> 4;
    const int lm   = lane & 15;
    const int w    = t >> 5;            // wave id 0..7
    const int b    = blockIdx.y;
    const int m0   = blockIdx.x * 16;

    // ---------------- Phase 1: WMMA src projection into LDS ----------------
    {
        const float* qrow = query + (size_t)(b * M_ + m0 + lm) * DQ_;
        const int h0 = w * 64;          // wave covers h in [h0, h0+64)

        v8f acc0 = {}, acc1 = {}, acc2 = {}, acc3 = {};
        for (int k = 0; k < DQ_; k += 4) {
            v2f a = *(const v2f*)(qrow + k + 2 * hl);   // reused for 4 tiles
            const float* bp0 = Wq + (size_t)(k + 2 * hl) * H_ + lm;
            const float* bp1 = bp0 + H_;
            v2f bb;
            bb.x = bp0[h0 +  0]; bb.y = bp1[h0 +  0];
            acc0 = __builtin_amdgcn_wmma_f32_16x16x4_f32(
                false, a, false, bb, (short)0, acc0, false, false);
            bb.x = bp0[h0 + 16]; bb.y = bp1[h0 + 16];
            acc1 = __builtin_amdgcn_wmma_f32_16x16x4_f32(
                false, a, false, bb, (short)0, acc1, false, false);
            bb.x = bp0[h0 + 32]; bb.y = bp1[h0 + 32];
            acc2 = __builtin_amdgcn_wmma_f32_16x16x4_f32(
                false, a, false, bb, (short)0, acc2, false, false);
            bb.x = bp0[h0 + 48]; bb.y = bp1[h0 + 48];
            acc3 = __builtin_amdgcn_wmma_f32_16x16x4_f32(
                false, a, false, bb, (short)0, acc3, false, false);
        }
#pragma unroll
        for (int v = 0; v < 8; ++v) {
            const int row = v + 8 * hl;     // D-matrix row mapping
            S[row][h0 +  0 + lm] = acc0[v];
            S[row][h0 + 16 + lm] = acc1[v];
            S[row][h0 + 32 + lm] = acc2[v];
            S[row][h0 + 48 + lm] = acc3[v];
        }
    }
    __syncthreads();

    // ---------------- Phase 2: fused relu-add-contract ----------------
    // thread t -> (mi = t&15, h-slice hs = t>>4 covering [hs*32, hs*32+32))
    const int mi = t & 15;
    const int hs = t >> 4;
    float ax = 0.f, ay = 0.f, az = 0.f, aw = 0.f;

    const float* trow_b = tgt + (size_t)b * N_ * H_;
    for (int n = 0; n < N_; ++n) {
        const float* trow = trow_b + (size_t)n * H_;
        const float* wrow = Wout + (size_t)n * H_ * 4;
#pragma unroll 8
        for (int j = 0; j < 32; ++j) {
            const int h = hs * 32 + j;
            float aval = S[mi][h] + trow[h];
            aval = fmaxf(aval, 0.0f);
            const float4 wv = *(const float4*)(wrow + (size_t)h * 4);
            ax = fmaf(aval, wv.x, ax);
            ay = fmaf(aval, wv.y, ay);
            az = fmaf(aval, wv.z, az);
            aw = fmaf(aval, wv.w, aw);
        }
    }
    red[mi][hs][0] = ax;
    red[mi][hs][1] = ay;
    red[mi][hs][2] = az;
    red[mi][hs][3] = aw;
    __syncthreads();

    if (t < 64) {
        const int mm = t >> 2, c = t & 3;
        float s = 0.f;
#pragma unroll
        for (int g = 0; g < 16; ++g) s += red[mm][g][c];
        out[(size_t)(b * M_ + m0 + mm) * 4 + c] = s;
    }
}

// ---------------------------------------------------------------------------
extern "C" void kernel_launch(void* const* d_in, const int* in_sizes, int n_in,
                              void* d_out, int out_size, void* d_ws, size_t ws_size,
                              hipStream_t stream) {
    const float* query = (const float*)d_in[0];   // [B,M,DQ]
    const float* mem   = (const float*)d_in[1];   // [B,N,DM]
    const float* Wq    = (const float*)d_in[2];   // [DQ,H]
    const float* Wm    = (const float*)d_in[3];   // [DM,H]
    const float* Wout  = (const float*)d_in[4];   // [N*H,4]
    float* out = (float*)d_out;                   // [B,M,4]
    float* tgt = (float*)d_ws;                    // [B,N,H] = 576 KB scratch

    // Kernel 1: tgt projection (576 wave-tiles, 8 waves/block)
    tgt_proj_kernel<<<72, 256, 0, stream>>>(mem, Wm, tgt);

    // Kernel 2: fused src-projection + broadcast-relu + Wout contraction
    fused_align_kernel<<<dim3(M_ / 16, B_), 256, 0, stream>>>(
        query, Wq, tgt, Wout, out);
}